// DE_NN_67912022884544
// MI455X (gfx1250) — compile-verified
//
#include <hip/hip_runtime.h>

#define NPOP 44
#define BSZ  400000

typedef float v2f __attribute__((ext_vector_type(2)));
typedef float v8f __attribute__((ext_vector_type(8)));

// Swap lanes 0-15 <-> 16-31 (group-of-32 swizzle, xor=0x10, and=0x1f)
__device__ __forceinline__ float swapx16(float v) {
    return __int_as_float(__builtin_amdgcn_ds_swizzle(__float_as_int(v), 0x401f));
}

// D(16x16 f32) = A(16x4 f32) * B(4x16 f32) + C
__device__ __forceinline__ v8f wmma4(v2f a, v2f b, v8f c) {
    return __builtin_amdgcn_wmma_f32_16x16x4_f32(false, a, false, b, (short)0, c,
                                                 false, false);
}

struct Net {
    v2f a1, a2, a3lo, a3hi, a4;
    float bb1_0, bb1_1;
    float bb2_0, bb2_1, bb2_2, bb2_3;
    float bb3_0, bb3_1;
};

// One 16-sample tile through the whole MLP. x carries the k=0 row of B in
// lanes 0-15 (lanes 16-31 / other k-rows are multiplied by zero A columns).
// Output: lanes 0-15 of the returned float hold the 16 results (bias4 not added).
__device__ __forceinline__ float mlp_forward(float x, const Net& N, v8f zc) {
    v2f b; b[0] = x; b[1] = x;
    v8f d = wmma4(N.a1, b, zc);                    // layer1: 1 -> 4
    v2f h1;
    h1[0] = fmaxf(d[0] + N.bb1_0, 0.f);            // rows {0,8} = n0,n2
    h1[1] = fmaxf(d[1] + N.bb1_1, 0.f);            // rows {1,9} = n1,n3
    d = wmma4(N.a2, h1, zc);                       // layer2: 4 -> 8
    v2f h2a, h2b;
    h2a[0] = fmaxf(d[0] + N.bb2_0, 0.f);           // n0 / n2
    h2a[1] = fmaxf(d[1] + N.bb2_1, 0.f);           // n1 / n3
    h2b[0] = fmaxf(d[2] + N.bb2_2, 0.f);           // n4 / n6
    h2b[1] = fmaxf(d[3] + N.bb2_3, 0.f);           // n5 / n7
    d = wmma4(N.a3lo, h2a, zc);                    // layer3: 8 -> 4 (K split)
    d = wmma4(N.a3hi, h2b, d);                     // accumulate second K half
    v2f h3;
    h3[0] = fmaxf(d[0] + N.bb3_0, 0.f);
    h3[1] = fmaxf(d[1] + N.bb3_1, 0.f);
    d = wmma4(N.a4, h3, zc);                       // layer4: 4 -> 1 (row 0)
    return d[0];
}

__global__ __launch_bounds__(256) void DE_NN_mlp44_wmma(
    const float* __restrict__ X,
    const float* __restrict__ lin1, const float* __restrict__ lin2,
    const float* __restrict__ lin3, const float* __restrict__ lin4,
    const float* __restrict__ b1,   const float* __restrict__ b2,
    const float* __restrict__ b3,   const float* __restrict__ b4,
    float* __restrict__ Out)
{
    const int l    = blockIdx.y;          // population
    const int lane = threadIdx.x & 31;    // wave32 lane
    const int row  = lane & 15;           // M (or N) index within half
    const int hi   = lane >> 4;           // 0: K even col, 1: K odd-half col

    // --- Build A matrices (16x4 f32 layout: v0 -> K=2*hi, v1 -> K=2*hi+1, M=row)
    // Neuron->row placement chosen so D of one layer == B of the next:
    //   4-wide layers: neurons {0,1,2,3} -> rows {0,1,8,9}
    //   8-wide layer : neurons {0,1,2,3,4,5,6,7} -> rows {0,1,8,9,2,3,10,11}
    const bool valid4 = ((row & 6) == 0);                 // rows 0,1,8,9
    const bool valid8 = ((row & 4) == 0);                 // rows 0-3, 8-11
    const int  n4 = (row & 1) | (((row >> 3) & 1) << 1);
    const int  n8 = (row & 1) | (((row >> 3) & 1) << 1) | (((row >> 1) & 1) << 2);

    Net N;
    N.a1[0]   = (valid4 && hi == 0) ? lin1[l * 4 + n4] : 0.f;   // K=0 column only
    N.a1[1]   = 0.f;
    N.a2[0]   = valid8 ? lin2[l * 32 + n8 * 4 + 2 * hi]     : 0.f;
    N.a2[1]   = valid8 ? lin2[l * 32 + n8 * 4 + 2 * hi + 1] : 0.f;
    N.a3lo[0] = valid4 ? lin3[l * 32 + n4 * 8 + 2 * hi]     : 0.f;
    N.a3lo[1] = valid4 ? lin3[l * 32 + n4 * 8 + 2 * hi + 1] : 0.f;
    N.a3hi[0] = valid4 ? lin3[l * 32 + n4 * 8 + 4 + 2 * hi] : 0.f;
    N.a3hi[1] = valid4 ? lin3[l * 32 + n4 * 8 + 5 + 2 * hi] : 0.f;
    N.a4[0]   = (row == 0) ? lin4[l * 4 + 2 * hi]     : 0.f;
    N.a4[1]   = (row == 0) ? lin4[l * 4 + 2 * hi + 1] : 0.f;

    // --- Per-lane bias values matching the D-row placement (hi half = rows 8+)
    N.bb1_0 = hi ? b1[l * 4 + 2] : b1[l * 4 + 0];
    N.bb1_1 = hi ? b1[l * 4 + 3] : b1[l * 4 + 1];
    N.bb2_0 = hi ? b2[l * 8 + 2] : b2[l * 8 + 0];
    N.bb2_1 = hi ? b2[l * 8 + 3] : b2[l * 8 + 1];
    N.bb2_2 = hi ? b2[l * 8 + 6] : b2[l * 8 + 4];
    N.bb2_3 = hi ? b2[l * 8 + 7] : b2[l * 8 + 5];
    N.bb3_0 = hi ? b3[l * 4 + 2] : b3[l * 4 + 0];
    N.bb3_1 = hi ? b3[l * 4 + 3] : b3[l * 4 + 1];
    const float bb4 = b4[l];                   // final bias is linear -> add at end

    const v8f zc = {};                          // shared zero accumulator

    const float* Xl = X   + (size_t)l * BSZ;
    float*       Ol = Out + (size_t)l * BSZ;

    const int gw = blockIdx.x * 8 + (threadIdx.x >> 5);   // global wave id
    const int nw = gridDim.x * 8;                         // total waves (per pop)

    // B = 400000 is a multiple of 128, so every chunk is full (EXEC stays all-1s).
    for (int base = gw * 128; base < BSZ; base += nw * 128) {
        const float4 xv = *reinterpret_cast<const float4*>(Xl + base + 4 * lane);
        float xin[4] = {xv.x, xv.y, xv.z, xv.w};
        float oo[4];
#pragma unroll
        for (int c = 0; c < 4; ++c) {
            const float x0 = xin[c];            // tile0: samples in lanes 0-15
            const float x1 = swapx16(x0);       // tile1: lanes 16-31 -> 0-15
            const float r0 = mlp_forward(x0, N, zc);
            const float r1 = mlp_forward(x1, N, zc);
            const float rs = swapx16(r1);       // tile1 results -> lanes 16-31
            oo[c] = (hi ? rs : r0) + bb4;
        }
        *reinterpret_cast<float4*>(Ol + base + 4 * lane) =
            make_float4(oo[0], oo[1], oo[2], oo[3]);
    }
}

extern "C" void kernel_launch(void* const* d_in, const int* in_sizes, int n_in,
                              void* d_out, int out_size, void* d_ws, size_t ws_size,
                              hipStream_t stream) {
    const float* X    = (const float*)d_in[0];
    const float* lin1 = (const float*)d_in[1];
    const float* lin2 = (const float*)d_in[2];
    const float* lin3 = (const float*)d_in[3];
    const float* lin4 = (const float*)d_in[4];
    const float* b1   = (const float*)d_in[5];
    const float* b2   = (const float*)d_in[6];
    const float* b3   = (const float*)d_in[7];
    const float* b4   = (const float*)d_in[8];
    float* out = (float*)d_out;

    dim3 grid(40, NPOP, 1);   // 40 blocks x 8 waves x ~10 grid-stride iters / pop
    dim3 block(256, 1, 1);
    DE_NN_mlp44_wmma<<<grid, block, 0, stream>>>(X, lin1, lin2, lin3, lin4,
                                                 b1, b2, b3, b4, out);
}